// LocalAggregation_71064528879968
// MI455X (gfx1250) — compile-verified
//
#include <hip/hip_runtime.h>
#include <cfloat>

#define RADIUS   0.2f
#define RAD2     (RADIUS * RADIUS)
#define NSAMPLE  32
#define BN_EPS   1e-5f
#define BATCH    4
#define CHAN     64
#define NPTS     8192
#define CIN      67      /* C + 3 */
#define CPAD     68      /* padded to multiple of 4 for 16x16x4 WMMA */

typedef __attribute__((ext_vector_type(2))) float v2f;
typedef __attribute__((ext_vector_type(8))) float v8f;

// ---------------------------------------------------------------------------
// Kernel 1: per-point KNN-32 with ball-query masking.
// One thread per query point; candidate coords tiled through LDS.
// ---------------------------------------------------------------------------
__global__ __launch_bounds__(256) void knn_kernel(const float* __restrict__ coor,
                                                  int* __restrict__ gidx) {
    __shared__ float sx[256], sy[256], sz[256];

    const int p = blockIdx.x * 256 + threadIdx.x;   // global point id
    const int b = p >> 13;                          // p / 8192
    const int n = p & (NPTS - 1);

    const float* cb = coor + (size_t)b * 3 * NPTS;  // (3, N) for this batch
    const float qx = cb[n];
    const float qy = cb[NPTS + n];
    const float qz = cb[2 * NPTS + n];

    float kd[NSAMPLE];
    int   ki[NSAMPLE];
#pragma unroll
    for (int i = 0; i < NSAMPLE; ++i) { kd[i] = FLT_MAX; ki[i] = 0; }

    for (int t = 0; t < NPTS; t += 256) {
        __syncthreads();
        sx[threadIdx.x] = cb[t + threadIdx.x];
        sy[threadIdx.x] = cb[NPTS + t + threadIdx.x];
        sz[threadIdx.x] = cb[2 * NPTS + t + threadIdx.x];
        __syncthreads();

        for (int i = 0; i < 256; ++i) {
            const float dx = sx[i] - qx;
            const float dy = sy[i] - qy;
            const float dz = sz[i] - qz;
            const float d  = dx * dx + dy * dy + dz * dz;
            if (d < kd[NSAMPLE - 1]) {
                int pos = NSAMPLE - 1;
                while (pos > 0 && kd[pos - 1] > d) {   // stable: equal keys keep order
                    kd[pos] = kd[pos - 1];
                    ki[pos] = ki[pos - 1];
                    --pos;
                }
                kd[pos] = d;
                ki[pos] = t + i;
            }
        }
    }

    // ball-query mask: neighbors outside radius fall back to the nearest (self)
    const int i0 = ki[0];
    int* outp = gidx + (size_t)p * NSAMPLE;
#pragma unroll
    for (int k = 0; k < NSAMPLE; ++k)
        outp[k] = (kd[k] > RAD2) ? i0 : ki[k];
}

// ---------------------------------------------------------------------------
// Kernel 2: fused gather -> fp32 WMMA GEMM (32x68 @ 68x64) -> BN -> ReLU ->
// max over neighborhood. One wave32 per point; 136 v_wmma_f32_16x16x4_f32
// per point. W zero-padded to 64x68 in LDS, BN folded to scale/bias in LDS.
// ---------------------------------------------------------------------------
__global__ __launch_bounds__(256) void agg_kernel(const float* __restrict__ coor,
                                                  const float* __restrict__ fea,
                                                  const float* __restrict__ W,
                                                  const float* __restrict__ gamma,
                                                  const float* __restrict__ beta,
                                                  const float* __restrict__ rmean,
                                                  const float* __restrict__ rvar,
                                                  const int* __restrict__ gidx,
                                                  float* __restrict__ out) {
    __shared__ float sW[CHAN * CPAD];   // 64 x 68, zero padded cols 67
    __shared__ float sScale[CHAN];
    __shared__ float sBias[CHAN];

    for (int i = threadIdx.x; i < CHAN * CPAD; i += 256) {
        const int d = i / CPAD;
        const int c = i - d * CPAD;
        sW[i] = (c < CIN) ? W[d * CIN + c] : 0.0f;
    }
    if (threadIdx.x < CHAN) {
        const float sc = gamma[threadIdx.x] * rsqrtf(rvar[threadIdx.x] + BN_EPS);
        sScale[threadIdx.x] = sc;
        sBias[threadIdx.x]  = beta[threadIdx.x] - rmean[threadIdx.x] * sc;
    }
    __syncthreads();

    const int lane = threadIdx.x & 31;
    const int wave = threadIdx.x >> 5;
    const int p    = blockIdx.x * 8 + wave;    // one point per wave
    const int b    = p >> 13;
    const int n    = p & (NPTS - 1);

    const float* fb = fea  + (size_t)b * CHAN * NPTS;  // (C, N)
    const float* cb = coor + (size_t)b * 3 * NPTS;     // (3, N)
    const float qx = cb[n], qy = cb[NPTS + n], qz = cb[2 * NPTS + n];
    const float invR = 1.0f / RADIUS;

    // neighbor index for this lane, then per-m-tile indices via shuffle
    const int myidx = gidx[(size_t)p * NSAMPLE + lane];
    const int lo    = lane & 15;
    const int hi    = lane >> 4;                 // 0: K=0/1 slots, 1: K=2/3 slots
    const int j0    = __shfl(myidx, lo);         // m-tile 0: neighbors 0..15
    const int j1    = __shfl(myidx, 16 + lo);    // m-tile 1: neighbors 16..31

    v8f acc[4][2];
#pragma unroll
    for (int dt = 0; dt < 4; ++dt) {
        v8f z = {};
        acc[dt][0] = z;
        acc[dt][1] = z;
    }

    // ---- main K-loop over the 64 feature channels (16 steps of 4) ----
    for (int s = 0; s < 16; ++s) {
        const int c = 4 * s + 2 * hi;            // A layout: lanes 0-15 -> K=c,c+1; 16-31 -> c+2,c+3
        v2f a0, a1;
        a0.x = fb[(size_t)c * NPTS + j0];
        a0.y = fb[(size_t)(c + 1) * NPTS + j0];
        a1.x = fb[(size_t)c * NPTS + j1];
        a1.y = fb[(size_t)(c + 1) * NPTS + j1];
#pragma unroll
        for (int dt = 0; dt < 4; ++dt) {
            const v2f bf = *(const v2f*)&sW[(16 * dt + lo) * CPAD + c];  // 8B aligned ds_load
            acc[dt][0] = __builtin_amdgcn_wmma_f32_16x16x4_f32(
                false, a0, false, bf, (short)0, acc[dt][0], false, false);
            acc[dt][1] = __builtin_amdgcn_wmma_f32_16x16x4_f32(
                false, a1, false, bf, (short)0, acc[dt][1], false, false);
        }
    }

    // ---- tail K-step: relative coords (channels 64..66) + zero pad (67) ----
    {
        v2f a0, a1;
        if (hi == 0) {
            a0.x = (cb[j0] - qx) * invR;            a0.y = (cb[NPTS + j0] - qy) * invR;
            a1.x = (cb[j1] - qx) * invR;            a1.y = (cb[NPTS + j1] - qy) * invR;
        } else {
            a0.x = (cb[2 * NPTS + j0] - qz) * invR; a0.y = 0.0f;
            a1.x = (cb[2 * NPTS + j1] - qz) * invR; a1.y = 0.0f;
        }
        const int c = CHAN + 2 * hi;                // 64 or 66
#pragma unroll
        for (int dt = 0; dt < 4; ++dt) {
            const v2f bf = *(const v2f*)&sW[(16 * dt + lo) * CPAD + c];
            acc[dt][0] = __builtin_amdgcn_wmma_f32_16x16x4_f32(
                false, a0, false, bf, (short)0, acc[dt][0], false, false);
            acc[dt][1] = __builtin_amdgcn_wmma_f32_16x16x4_f32(
                false, a1, false, bf, (short)0, acc[dt][1], false, false);
        }
    }

    // ---- epilogue: BN affine per element, max over 32 neighbors, ReLU ----
    // C/D layout: VGPR r, lanes 0-15 -> row r, lanes 16-31 -> row r+8; col = lane&15.
#pragma unroll
    for (int dt = 0; dt < 4; ++dt) {
        const int   d  = 16 * dt + lo;
        const float sc = sScale[d];
        const float bi = sBias[d];
        float m = -FLT_MAX;
#pragma unroll
        for (int r = 0; r < 8; ++r) {
            m = fmaxf(m, fmaf(acc[dt][0][r], sc, bi));
            m = fmaxf(m, fmaf(acc[dt][1][r], sc, bi));
        }
        m = fmaxf(m, __shfl_xor(m, 16));   // combine rows 0-7/16-23 with 8-15/24-31
        m = fmaxf(m, 0.0f);                // ReLU (monotonic, safe after max)
        if (lane < 16)
            out[(size_t)b * CHAN * NPTS + (size_t)d * NPTS + n] = m;
    }
}

// ---------------------------------------------------------------------------
extern "C" void kernel_launch(void* const* d_in, const int* in_sizes, int n_in,
                              void* d_out, int out_size, void* d_ws, size_t ws_size,
                              hipStream_t stream) {
    const float* coor  = (const float*)d_in[0];   // (B, 3, N)
    const float* fea   = (const float*)d_in[1];   // (B, C, N)
    const float* W     = (const float*)d_in[2];   // (C, C+3)
    const float* gamma = (const float*)d_in[3];
    const float* beta  = (const float*)d_in[4];
    const float* rmean = (const float*)d_in[5];
    const float* rvar  = (const float*)d_in[6];
    float* out = (float*)d_out;                   // (B, C, N)
    int*   gidx = (int*)d_ws;                     // (B*N, 32) neighbor indices, 4 MB

    const int total_pts = BATCH * NPTS;           // 32768
    knn_kernel<<<total_pts / 256, 256, 0, stream>>>(coor, gidx);
    agg_kernel<<<total_pts / 8, 256, 0, stream>>>(coor, fea, W, gamma, beta,
                                                  rmean, rvar, gidx, out);
}